// NeuraLogicHelperLayer_17291538334287
// MI455X (gfx1250) — compile-verified
//
#include <hip/hip_runtime.h>
#include <hip/hip_bf16.h>
#include <stdint.h>

// -----------------------------------------------------------------------------
// Graph message-passing:  out = old_x + new_x
//   old_x = x with target rows zeroed
//   msg   = x[u] * weights[w_idx]   (elementwise, FEAT=64)
//   agg   = segment_sum(msg, v); deg = segment_sum(1, v)
//   new_x = deg>0 ? tanh(agg) : x
// Memory/atomic-bound (~2.8 GB @ 23.3 TB/s). No GEMM content -> WMMA N/A.
// CDNA5 paths: global_load_async_to_lds_b128 staging of the 256 KB weight
// table (fits in 320 KB/WGP LDS), s_wait_asynccnt, wave32 edge-per-wave,
// explicit hardware global_atomic_add_f32 (non-returning, scope DEV).
// -----------------------------------------------------------------------------

#define FEAT 64
#define FEAT2 32  // float2 elements per row

__device__ __forceinline__ void hw_atomic_add_f32(float* p, float v) {
  // Non-returning f32 atomic add executed by the L2 atomic units.
  asm volatile("global_atomic_add_f32 %0, %1, off scope:SCOPE_DEV"
               :
               : "v"(p), "v"(v)
               : "memory");
}

// ---- Hot kernel first (so the disasm snippet shows it): edge scatter --------
// Dynamic LDS = nWeights*FEAT*4 bytes (256 KB for 1024 weights).
extern __shared__ float2 s_w2[];

__global__ void k_edges(const float* __restrict__ x,
                        const float* __restrict__ weights,
                        const int* __restrict__ u,
                        const int* __restrict__ v,
                        const int* __restrict__ w_idx,
                        float* __restrict__ out,      // agg accumulator
                        unsigned* __restrict__ deg,
                        int nEdges, int nWeights) {
  // ---- Stage the whole weight table into LDS via CDNA5 async copies ----
  {
    const char* gbase = (const char*)weights;
    char* lbase = (char*)&s_w2[0];
    int nChunks = nWeights * FEAT * 4 / 16;  // 16 B per b128 transfer
    for (int c = threadIdx.x; c < nChunks; c += blockDim.x) {
      unsigned lds_off = (unsigned)(unsigned long long)(lbase + (size_t)c * 16);
      const char* gaddr = gbase + (size_t)c * 16;
      // ASYNCcnt-tracked DMA-style copy, no VGPR data movement (ISA 08 §4.2)
      asm volatile("global_load_async_to_lds_b128 %0, %1, off"
                   :
                   : "v"(lds_off), "v"(gaddr)
                   : "memory");
    }
    asm volatile("s_wait_asynccnt 0" ::: "memory");  // my loads landed in LDS
    __syncthreads();                                  // everyone's did
  }

  const float2* x2 = (const float2*)x;
  int lane = threadIdx.x & 31;                      // wave32
  int waveId = blockIdx.x * (blockDim.x >> 5) + (threadIdx.x >> 5);
  int nWaves = gridDim.x * (blockDim.x >> 5);

  // Each wave consumes 32 edges per batch: coalesced index loads, then
  // broadcast each edge's (u,v,w) to all lanes via shfl.
  for (int base = waveId * 32; base < nEdges; base += nWaves * 32) {
    int e = base + lane;
    int ue = 0, ve = 0, we = 0;
    if (e < nEdges) { ue = u[e]; ve = v[e]; we = w_idx[e]; }
    int cnt = nEdges - base; if (cnt > 32) cnt = 32;

    for (int j = 0; j < cnt; ++j) {
      int uj = __shfl(ue, j);
      int vj = __shfl(ve, j);
      int wj = __shfl(we, j);
      // 32 lanes x float2 = one coalesced 256 B row
      float2 xv = x2[(size_t)uj * FEAT2 + lane];
      float2 wv = s_w2[(size_t)wj * FEAT2 + lane];
      float mx = xv.x * wv.x;
      float my = xv.y * wv.y;
      float* dst = out + (size_t)vj * FEAT + lane * 2;
      hw_atomic_add_f32(dst + 0, mx);   // segment-sum into agg (d_out)
      hw_atomic_add_f32(dst + 1, my);
      if (lane == 0) atomicAdd(&deg[vj], 1u);
    }
  }
}

// ---- Zero d_out (agg accumulator), deg, flags -------------------------------
__global__ void k_init(float4* __restrict__ out4, long long nOut4,
                       unsigned* __restrict__ deg, int nNodes,
                       unsigned* __restrict__ flags32, int nFlags32) {
  long long stride = (long long)gridDim.x * blockDim.x;
  long long tid = (long long)blockIdx.x * blockDim.x + threadIdx.x;
  float4 z = make_float4(0.f, 0.f, 0.f, 0.f);
  for (long long i = tid; i < nOut4; i += stride) out4[i] = z;
  for (long long i = tid; i < nNodes; i += stride) deg[i] = 0u;
  for (long long i = tid; i < nFlags32; i += stride) flags32[i] = 0u;
}

// ---- Mark target nodes ------------------------------------------------------
__global__ void k_mark(const int* __restrict__ targets, int nTargets,
                       unsigned char* __restrict__ flags) {
  int i = blockIdx.x * blockDim.x + threadIdx.x;
  if (i < nTargets) {
    int t = targets[i];
    flags[t] = 1;  // idempotent; duplicate targets race benignly
  }
}

// ---- Finalize:  out = old_x + (deg>0 ? tanh(agg) : x),  float4-wide ---------
__global__ void k_final(const float* __restrict__ x,
                        float* __restrict__ out,
                        const unsigned* __restrict__ deg,
                        const unsigned char* __restrict__ flags,
                        long long nElems4) {   // nNodes * FEAT/4
  long long i = (long long)blockIdx.x * blockDim.x + threadIdx.x;
  if (i >= nElems4) return;
  long long n = i >> 4;              // node (i / (FEAT/4))
  const float4* x4 = (const float4*)x;
  float4* o4 = (float4*)out;

  float4 a = o4[i];                  // accumulated agg
  float4 xv = x4[i];
  bool hasMsg = deg[n] > 0u;
  float4 nx;
  nx.x = hasMsg ? tanhf(a.x) : xv.x;
  nx.y = hasMsg ? tanhf(a.y) : xv.y;
  nx.z = hasMsg ? tanhf(a.z) : xv.z;
  nx.w = hasMsg ? tanhf(a.w) : xv.w;
  float s = (flags[n] != 0) ? 0.f : 1.f;   // old_x contribution
  o4[i] = make_float4(s * xv.x + nx.x, s * xv.y + nx.y,
                      s * xv.z + nx.z, s * xv.w + nx.w);
}

// -----------------------------------------------------------------------------
extern "C" void kernel_launch(void* const* d_in, const int* in_sizes, int n_in,
                              void* d_out, int out_size, void* d_ws, size_t ws_size,
                              hipStream_t stream) {
  const float* x       = (const float*)d_in[0];
  const float* weights = (const float*)d_in[1];
  const int*   u       = (const int*)d_in[2];
  const int*   v       = (const int*)d_in[3];
  const int*   w_idx   = (const int*)d_in[4];
  const int*   targets = (const int*)d_in[5];
  float* out = (float*)d_out;

  int nNodes   = in_sizes[0] / FEAT;
  int nWeights = in_sizes[1] / FEAT;
  int nEdges   = in_sizes[2];
  int nTargets = in_sizes[5];

  // Workspace layout: [deg: nNodes u32][flags: nNodes u8]
  unsigned* deg = (unsigned*)d_ws;
  unsigned char* flags = (unsigned char*)d_ws + (size_t)nNodes * 4;

  // 1) zero agg accumulator (d_out), deg, flags
  long long nOut4 = (long long)out_size / 4;
  k_init<<<8192, 256, 0, stream>>>((float4*)out, nOut4, deg, nNodes,
                                   (unsigned*)flags, nNodes / 4);

  // 2) mark target nodes
  k_mark<<<(nTargets + 255) / 256, 256, 0, stream>>>(targets, nTargets, flags);

  // 3) edge scatter; 256 KB dynamic LDS holds whole weight table per workgroup
  size_t ldsBytes = (size_t)nWeights * FEAT * 4;
  k_edges<<<512, 1024, ldsBytes, stream>>>(x, weights, u, v, w_idx, out, deg,
                                           nEdges, nWeights);

  // 4) finalize (float4-wide over all 64M outputs)
  long long nElems4 = (long long)out_size / 4;
  int blocks = (int)((nElems4 + 255) / 256);
  k_final<<<blocks, 256, 0, stream>>>(x, out, deg, flags, nElems4);
}